// DecoderHeadVDP_50388556316851
// MI455X (gfx1250) — compile-verified
//
#include <hip/hip_runtime.h>

// ---------------------------------------------------------------------------
// DecoderHeadVDP on MI455X (gfx1250, wave32, WMMA + TDM)
// B=4, S=1024, D=1024, H=16, DH=64
// fp32 throughout via V_WMMA_F32_16X16X4_F32 (variance path is precision
// sensitive: p^2 / p^3 terms on 0.1-scale variances). Tile staging into LDS
// uses the Tensor Data Mover (tensor_load_to_lds) with descriptor padding
// reproducing the bank-conflict-free row stride of 65 DWORDs.
// ---------------------------------------------------------------------------

#define BATCH 4
#define SEQ   1024
#define DIM   1024
#define HEADS 16
#define DHEAD 64
#define RSQD  (1.0f / 32.0f)     // 1/sqrt(D)
#define RVD   (1.0f / 1024.0f)   // 1/D

typedef __attribute__((ext_vector_type(2))) float        v2f;
typedef __attribute__((ext_vector_type(8))) float        v8f;
typedef __attribute__((ext_vector_type(4))) unsigned int v4u;
typedef __attribute__((ext_vector_type(8))) int          v8i;
typedef __attribute__((ext_vector_type(4))) int          v4i;

#if defined(__has_builtin)
#if __has_builtin(__builtin_amdgcn_tensor_load_to_lds)
#define USE_TDM 1
#endif
#endif

static __device__ __forceinline__ v8f wmma4(v2f a, v2f b, v8f c) {
  // D = A(16x4) * B(4x16) + C, fp32
  return __builtin_amdgcn_wmma_f32_16x16x4_f32(false, a, false, b, (short)0, c,
                                               false, false);
}

// reductions across the 16-lane row group (wave32; xor masks <=8 stay in-half)
static __device__ __forceinline__ float rowmax16(float v) {
  v = fmaxf(v, __shfl_xor(v, 1));
  v = fmaxf(v, __shfl_xor(v, 2));
  v = fmaxf(v, __shfl_xor(v, 4));
  v = fmaxf(v, __shfl_xor(v, 8));
  return v;
}
static __device__ __forceinline__ float rowsum16(float v) {
  v += __shfl_xor(v, 1);
  v += __shfl_xor(v, 2);
  v += __shfl_xor(v, 4);
  v += __shfl_xor(v, 8);
  return v;
}

#ifdef USE_TDM
// ---------------------------------------------------------------------------
// TDM: async DMA of a 16-row x 64-col f32 tile, global -> LDS.
//   src             : global address of tile start (row-major, row pitch
//                     `pitch_elems` floats)
//   lds_byte_off    : LDS byte offset of destination
// D# pad feature inserts 1 DWORD after every 64 DWORDs -> LDS row stride 65,
// matching the padded tiles used by the WMMA operand reads.
// clang-23 / therock toolchain: 6-arg builtin
//   (v4u g0, v8i g1, v4i g2, v4i g3, v8i g4, i32 cpol)
// ---------------------------------------------------------------------------
static __device__ __forceinline__ void tdm_load_tile16x64(
    const float* src, unsigned int lds_byte_off, unsigned int pitch_elems) {
  const unsigned long long ga = (unsigned long long)(uintptr_t)src;
  v4u g0;
  g0.x = 1u;                                      // count=1 (valid), user mode
  g0.y = lds_byte_off;                            // lds_addr
  g0.z = (unsigned int)(ga & 0xffffffffu);        // global_addr[31:0]
  g0.w = (unsigned int)((ga >> 32) & 0x01ffffffu) // global_addr[56:32]
         | (2u << 30);                            // type = 2 ("image")
  v8i g1;
  g1[0] = (int)((2u << 16)        // data_size = 4 bytes
                | (1u << 20)      // pad_enable
                | (5u << 22));    // pad_interval: 64 DWORDs; pad_amount=0 -> 1 DWORD
  g1[1] = (int)(64u << 16);       // tensor_dim0[15:0] = 64
  g1[2] = (int)(16u << 16);       // tensor_dim0 hi=0 | tensor_dim1[15:0] = 16
  g1[3] = (int)(64u << 16);       // tensor_dim1 hi=0 | tile_dim0 = 64
  g1[4] = (int)16u;               // tile_dim1 = 16 | tile_dim2 = 0
  g1[5] = (int)pitch_elems;       // tensor_dim0_stride[31:0]
  g1[6] = 0;                      // stride hi | tensor_dim1_stride lo
  g1[7] = 0;
  const v4i z4 = {0, 0, 0, 0};                    // groups 2/3 unused (2D)
  const v8i z8 = {0, 0, 0, 0, 0, 0, 0, 0};
  __builtin_amdgcn_tensor_load_to_lds(g0, g1, z4, z4, z8, 0);
}
#endif

// ---------------------------------------------------------------------------
// Kernel 0: fold W2 = Wq_var + Wq_mu^2   (D*D elements)
// ---------------------------------------------------------------------------
__global__ void vdp_prep_w2(const float* __restrict__ wmu,
                            const float* __restrict__ wvar,
                            float* __restrict__ w2) {
  int i = blockIdx.x * blockDim.x + threadIdx.x;
  if (i < DIM * DIM) {
    float m = wmu[i];
    w2[i] = wvar[i] + m * m;
  }
}

// ---------------------------------------------------------------------------
// Kernel 1: projection.  q = x @ Wmu^T ; var_q = var_x @ W2^T + x^2 @ Wvar^T
// One wave per 16x16 output tile, K=1024 -> 256 wmma steps, 3 chains
// (variance chains share one accumulator). Also emits q^2 for the score GEMM.
// Weights are L2-resident (4MB each on a 192MB L2), so direct global loads.
// ---------------------------------------------------------------------------
__global__ void vdp_proj(const float* __restrict__ x,
                         const float* __restrict__ var_x,
                         const float* __restrict__ wmu,
                         const float* __restrict__ wvar,
                         const float* __restrict__ w2,
                         float* __restrict__ q,
                         float* __restrict__ var_q,
                         float* __restrict__ qsq) {
  const int lane = threadIdx.x & 31;
  const int wave = threadIdx.x >> 5;
  const int half = lane >> 4;   // 0/1
  const int l16  = lane & 15;

  const int tile = blockIdx.x * 8 + wave;        // 16384 tiles total
  const int m0 = (tile >> 6) << 4;               // (B*S)/16 = 256 row tiles
  const int n0 = (tile & 63) << 4;               // D/16 = 64 col tiles

  const float* xr  = x     + (size_t)(m0 + l16) * DIM;
  const float* vxr = var_x + (size_t)(m0 + l16) * DIM;
  const float* wmr = wmu   + (size_t)(n0 + l16) * DIM;
  const float* w2r = w2    + (size_t)(n0 + l16) * DIM;
  const float* wvr = wvar  + (size_t)(n0 + l16) * DIM;

  v8f accQ = {};
  v8f accV = {};
  const int kb = half * 2;

  for (int t = 0; t < DIM / 4; ++t) {
    const int kk = t * 4 + kb;
    v2f ax;  ax.x  = xr[kk];   ax.y  = xr[kk + 1];
    v2f avx; avx.x = vxr[kk];  avx.y = vxr[kk + 1];
    v2f ax2; ax2.x = ax.x * ax.x; ax2.y = ax.y * ax.y;
    v2f bm;  bm.x  = wmr[kk];  bm.y  = wmr[kk + 1];
    v2f b2;  b2.x  = w2r[kk];  b2.y  = w2r[kk + 1];
    v2f bv;  bv.x  = wvr[kk];  bv.y  = wvr[kk + 1];
    accQ = wmma4(ax,  bm, accQ);
    accV = wmma4(avx, b2, accV);
    accV = wmma4(ax2, bv, accV);
  }

#pragma unroll
  for (int r = 0; r < 8; ++r) {
    const int m = m0 + r + 8 * half;
    const int n = n0 + l16;
    const size_t gi = (size_t)m * DIM + n;
    const float qm = accQ[r];
    q[gi]     = qm;
    qsq[gi]   = qm * qm;
    var_q[gi] = accV[r];
  }
}

// ---------------------------------------------------------------------------
// Kernel 2: fused causal flash-attention with variance propagation.
// One wave per (b, h, 16-row block). Never materializes the SxS scores.
// Tile staging via TDM (tensor_load_to_lds), synced with TENSORcnt.
// ---------------------------------------------------------------------------
#define TS 65   // padded row stride for 16x64 LDS tiles (bank-conflict free)
#define PS 17   // padded row stride for 16x16 LDS tiles

__global__ __launch_bounds__(32)
void vdp_flash(const float* __restrict__ q,
               const float* __restrict__ var_q,
               const float* __restrict__ qsq,
               const float* __restrict__ kk_,
               const float* __restrict__ var_k,
               const float* __restrict__ vv_,
               const float* __restrict__ var_v,
               const float* __restrict__ x,
               float* __restrict__ out_mean,
               float* __restrict__ out_var) {
  __shared__ float lq[16 * TS], lvq[16 * TS], lq2[16 * TS];
  __shared__ float lk[16 * TS], lkv[16 * TS];
  __shared__ float lv[16 * TS], lvv[16 * TS];
  __shared__ float lp[4][16 * PS];

  const int lane = threadIdx.x & 31;
  const int half = lane >> 4;
  const int l16  = lane & 15;

  const int iblk = blockIdx.x & 63;        // S/16 row blocks
  const int bh   = blockIdx.x >> 6;
  const int h    = bh & (HEADS - 1);
  const int b    = bh >> 4;
  const int i0   = iblk * 16;

  const size_t khBase = (size_t)bh * SEQ * DHEAD;

  // ---- stage the q / var_q / q^2 row tiles (16 x 64, pitch DIM) once ----
#ifdef USE_TDM
  {
    const size_t qOff = (size_t)(b * SEQ + i0) * DIM + h * DHEAD;
    tdm_load_tile16x64(q     + qOff, (unsigned int)(uintptr_t)(&lq [0]), DIM);
    tdm_load_tile16x64(var_q + qOff, (unsigned int)(uintptr_t)(&lvq[0]), DIM);
    tdm_load_tile16x64(qsq   + qOff, (unsigned int)(uintptr_t)(&lq2[0]), DIM);
    __builtin_amdgcn_s_wait_tensorcnt((short)0);
  }
#else
  for (int idx = lane; idx < 16 * DHEAD; idx += 32) {
    const int rr = idx >> 6, cc = idx & 63;
    const size_t gi = (size_t)(b * SEQ + i0 + rr) * DIM + h * DHEAD + cc;
    lq [rr * TS + cc] = q[gi];
    lvq[rr * TS + cc] = var_q[gi];
    lq2[rr * TS + cc] = qsq[gi];
  }
#endif
  __syncthreads();   // single-wave WG: near-free, forces LDS visibility

  float mrun[8], lrun[8], s2run[8];
  v8f accA[4], accB1[4], accB2[4], accB3[4], accB4[4];
#pragma unroll
  for (int r = 0; r < 8; ++r) { mrun[r] = -1e30f; lrun[r] = 0.f; s2run[r] = 0.f; }
#pragma unroll
  for (int nt = 0; nt < 4; ++nt) {
    accA[nt] = (v8f){}; accB1[nt] = (v8f){}; accB2[nt] = (v8f){};
    accB3[nt] = (v8f){}; accB4[nt] = (v8f){};
  }

  for (int j0 = 0; j0 <= i0; j0 += 16) {
    // ---- stage k / var_k / v / var_v column tiles (16 x 64, pitch DHEAD) ----
#ifdef USE_TDM
    {
      // make sure prior DS reads of these tiles have drained before the DMA
      // overwrites them (TDM writes are not ordered against DS ops)
      asm volatile("s_wait_dscnt 0x0" ::: "memory");
      const size_t kOff = khBase + (size_t)j0 * DHEAD;
      tdm_load_tile16x64(kk_   + kOff, (unsigned int)(uintptr_t)(&lk [0]), DHEAD);
      tdm_load_tile16x64(var_k + kOff, (unsigned int)(uintptr_t)(&lkv[0]), DHEAD);
      tdm_load_tile16x64(vv_   + kOff, (unsigned int)(uintptr_t)(&lv [0]), DHEAD);
      tdm_load_tile16x64(var_v + kOff, (unsigned int)(uintptr_t)(&lvv[0]), DHEAD);
      __builtin_amdgcn_s_wait_tensorcnt((short)0);
    }
#else
    for (int idx = lane; idx < 16 * DHEAD; idx += 32) {
      const int rr = idx >> 6, cc = idx & 63;
      const size_t gi = khBase + (size_t)(j0 + rr) * DHEAD + cc;
      lk [rr * TS + cc] = kk_[gi];
      lkv[rr * TS + cc] = var_k[gi];
      lv [rr * TS + cc] = vv_[gi];
      lvv[rr * TS + cc] = var_v[gi];
    }
#endif
    __syncthreads();

    // ---- score GEMMs: mean = q k^T ; var = var_q (var_k+k^2)^T + q^2 var_k^T
    v8f sm = {}, sv = {};
    const int kb = half * 2;
#pragma unroll
    for (int t = 0; t < DHEAD / 4; ++t) {
      const int kkx = t * 4 + kb;
      const int qi = l16 * TS + kkx;
      v2f aq;  aq.x  = lq [qi]; aq.y  = lq [qi + 1];
      v2f avq; avq.x = lvq[qi]; avq.y = lvq[qi + 1];
      v2f aq2; aq2.x = lq2[qi]; aq2.y = lq2[qi + 1];
      v2f bk;  bk.x  = lk [qi]; bk.y  = lk [qi + 1];     // B = k^T: row n=l16
      v2f bkv; bkv.x = lkv[qi]; bkv.y = lkv[qi + 1];
      v2f bw;  bw.x = bkv.x + bk.x * bk.x; bw.y = bkv.y + bk.y * bk.y;
      sm = wmma4(aq,  bk,  sm);
      sv = wmma4(avq, bw,  sv);
      sv = wmma4(aq2, bkv, sv);
    }

    // ---- online VDP softmax update ----
    float f[8];
#pragma unroll
    for (int r = 0; r < 8; ++r) {
      const int m = i0 + r + 8 * half;          // global row
      const int n = j0 + l16;                   // global col
      const bool ok = (n <= m);
      float a  = ok ? sm[r] * RSQD : -1e30f;
      float va = sv[r] * RVD;
      const float rmax = rowmax16(a);
      const float mnew = fmaxf(mrun[r], rmax);
      f[r] = __expf(mrun[r] - mnew);
      mrun[r] = mnew;
      const float p   = ok ? __expf(a - mnew) : 0.0f;
      const float p2  = p * p;
      const float p2v = p2 * va;
      lrun[r]  = lrun[r]  * f[r]        + rowsum16(p);
      s2run[r] = s2run[r] * f[r] * f[r] + rowsum16(p2v);
      const int li = (r + 8 * half) * PS + l16; // store in (M,N) layout
      lp[0][li] = p;
      lp[1][li] = p2;
      lp[2][li] = p2v;
      lp[3][li] = p2v * p;
    }
    // rescale output accumulators (A ~ f, B1/B2/B4 ~ f^2, B3 ~ f^3)
#pragma unroll
    for (int r = 0; r < 8; ++r) {
      const float fr = f[r], f2 = fr * fr, f3 = f2 * fr;
#pragma unroll
      for (int nt = 0; nt < 4; ++nt) {
        accA [nt][r] *= fr;
        accB1[nt][r] *= f2;
        accB2[nt][r] *= f2;
        accB4[nt][r] *= f2;
        accB3[nt][r] *= f3;
      }
    }
    __syncthreads();

    // ---- output GEMMs: A += P1@v ; B1 += P2@W ; B2 += P3@W ; B3 += P4@W ;
    // ----               B4 += P2@var_v   (W = var_v + v^2)
#pragma unroll
    for (int t = 0; t < 4; ++t) {
      const int kkx = t * 4 + kb;
      const int pi = l16 * PS + kkx;            // A layout: m=l16, k=kkx+v
      v2f a1; a1.x = lp[0][pi]; a1.y = lp[0][pi + 1];
      v2f a2; a2.x = lp[1][pi]; a2.y = lp[1][pi + 1];
      v2f a3; a3.x = lp[2][pi]; a3.y = lp[2][pi + 1];
      v2f a4; a4.x = lp[3][pi]; a4.y = lp[3][pi + 1];
#pragma unroll
      for (int nt = 0; nt < 4; ++nt) {
        const int col = nt * 16 + l16;
        const float bv0 = lv [kkx * TS + col];
        const float bv1 = lv [(kkx + 1) * TS + col];
        const float bq0 = lvv[kkx * TS + col];
        const float bq1 = lvv[(kkx + 1) * TS + col];
        v2f Bv;  Bv.x  = bv0;             Bv.y  = bv1;
        v2f Bq;  Bq.x  = bq0;             Bq.y  = bq1;
        v2f Bw;  Bw.x  = bq0 + bv0 * bv0; Bw.y  = bq1 + bv1 * bv1;
        accA [nt] = wmma4(a1, Bv, accA [nt]);
        accB1[nt] = wmma4(a2, Bw, accB1[nt]);
        accB2[nt] = wmma4(a3, Bw, accB2[nt]);
        accB3[nt] = wmma4(a4, Bw, accB3[nt]);
        accB4[nt] = wmma4(a2, Bq, accB4[nt]);
      }
    }
    __syncthreads();
  }

  // ---- finalize: mean = x + A/l ;  var = s*B1/l^2 + (B2+B4)/l^2 - 2 B3/l^3
#pragma unroll
  for (int r = 0; r < 8; ++r) {
    const float inv  = 1.0f / lrun[r];
    const float inv2 = inv * inv;
    const float s    = s2run[r] * inv2;
    const int m = i0 + r + 8 * half;
#pragma unroll
    for (int nt = 0; nt < 4; ++nt) {
      const int col = h * DHEAD + nt * 16 + l16;
      const size_t gi = (size_t)(b * SEQ + m) * DIM + col;
      out_mean[gi] = x[gi] + accA[nt][r] * inv;
      out_var[gi]  = (s * accB1[nt][r] + accB2[nt][r] + accB4[nt][r]) * inv2
                     - 2.0f * accB3[nt][r] * inv2 * inv;
    }
  }
}

// ---------------------------------------------------------------------------
extern "C" void kernel_launch(void* const* d_in, const int* in_sizes, int n_in,
                              void* d_out, int out_size, void* d_ws, size_t ws_size,
                              hipStream_t stream) {
  const float* x     = (const float*)d_in[0];
  const float* var_x = (const float*)d_in[1];
  const float* k     = (const float*)d_in[2];
  const float* var_k = (const float*)d_in[3];
  const float* v     = (const float*)d_in[4];
  const float* var_v = (const float*)d_in[5];
  const float* wmu   = (const float*)d_in[6];
  const float* wvar  = (const float*)d_in[7];

  float* out = (float*)d_out;               // [mean | var], each B*S*D
  float* ws  = (float*)d_ws;
  const size_t NT = (size_t)BATCH * SEQ * DIM;   // 4M floats
  float* q   = ws;
  float* vq  = ws + NT;
  float* q2  = ws + 2 * NT;
  float* w2  = ws + 3 * NT;                 // D*D = 1M floats

  // 0) fold Wq_var + Wq_mu^2
  vdp_prep_w2<<<(DIM * DIM + 255) / 256, 256, 0, stream>>>(wmu, wvar, w2);

  // 1) projection: 16384 16x16 tiles, 8 waves/block
  vdp_proj<<<2048, 256, 0, stream>>>(x, var_x, wmu, wvar, w2, q, vq, q2);

  // 2) fused causal flash VDP attention + residual, 1 wave per 16-row block
  vdp_flash<<<BATCH * HEADS * (SEQ / 16), 32, 0, stream>>>(
      q, vq, q2, k, var_k, v, var_v, x, out, out + NT);
}